// GlobalClusterBlock_1434519077363
// MI455X (gfx1250) — compile-verified
//
#include <hip/hip_runtime.h>
#include <math.h>

// ---------------------------------------------------------------------------
// MI455X (gfx1250) implementation. All GEMMs use V_WMMA_F32_16X16X4_F32
// (wave32, full fp32 precision). The pipeline is memory-bound (~107 GFLOP but
// only a few hundred MB of HBM traffic at 23.3 TB/s), so we keep fp32 and
// spend effort on fusion + data movement: reshape/scatter GEMM epilogues,
// GELU in the GEMM epilogue, conv bias+LayerNorm+residual fused in the conv
// kernel, and LDS tile staging via the CDNA5 async copy path
// (GLOBAL_LOAD_ASYNC_TO_LDS_B128 / ASYNCcnt) instead of VGPR round-trips.
// ---------------------------------------------------------------------------

typedef __attribute__((ext_vector_type(2))) float v2f;
typedef __attribute__((ext_vector_type(8))) float v8f;

__device__ __forceinline__ v8f zero8() {
  v8f z;
#pragma unroll
  for (int i = 0; i < 8; ++i) z[i] = 0.f;
  return z;
}

// D = A(16x4) * B(4x16) + C, fp32.
// A frag: lane = (k>=2?16:0)+m, vgpr = k&1  -> a.x=A[m][hi*2], a.y=A[m][hi*2+1]
// B frag: lane = (k>=2?16:0)+n, vgpr = k&1
// C/D   : vgpr i, lanes 0-15 -> (row i, col lane); lanes 16-31 -> (row i+8, col lane-16)
__device__ __forceinline__ v8f wmma4(v2f a, v2f b, v8f c) {
  return __builtin_amdgcn_wmma_f32_16x16x4_f32(false, a, false, b, (short)0, c,
                                               false, false);
}

// --- CDNA5 async copy: global -> LDS, 16B per lane, tracked by ASYNCcnt ----
// Generic pointers to __shared__ carry the wave-relative LDS byte offset in
// their low 32 bits (addrspacecast AS3->generic is {shared_base_hi, offset}),
// which is exactly what the async-load VDST operand wants.
__device__ __forceinline__ unsigned lds_addr(const void* p) {
  return (unsigned)(uintptr_t)p;
}
__device__ __forceinline__ void async_b128(unsigned lds_byte, int voff_byte,
                                           const void* sbase) {
  asm volatile("global_load_async_to_lds_b128 %0, %1, %2"
               :
               : "v"(lds_byte), "v"(voff_byte), "s"(sbase)
               : "memory");
}
__device__ __forceinline__ void async_wait0() {
  asm volatile("s_wait_asynccnt 0" ::: "memory");
}

// ---------------------------------------------------------------------------
// Generic 64x64-tile GEMM:  out = A @ W^T + bias, W row-major [N][K].
// A is split into two column-blocks (A0: k<K0, A1: k>=K0) so the lin GEMM can
// read the virtual concat [x0 | new] without materializing it. K0 and all
// k-chunks are 64-aligned, so the A source base is uniform per chunk.
// Epilogue modes:
//   0: scatter to xp per-head layout  (n,l,8,32)->(n*8,l,32)
//   1: scatter to cp/cv split         (n,s,8,64)->(n*8,s,32)+(n*8,s,32)
//   2: plain store (row-major, stride N)
//   3: exact GELU then store
// ---------------------------------------------------------------------------
__device__ __forceinline__ void gemm_store(int mode, int row, int col,
                                           float val,
                                           const float* __restrict__ bias,
                                           float* __restrict__ out,
                                           float* __restrict__ out2, int N) {
  val += bias[col];
  if (mode == 0) {
    int n = row >> 12, l = row & 4095;
    int f = col >> 5, j = col & 31;
    out[((size_t)((n << 3) + f) * 4096 + l) * 32 + j] = val;
  } else if (mode == 1) {
    int n = row >> 8, si = row & 255;
    int f = col >> 6, j = col & 63;
    size_t base = ((size_t)((n << 3) + f) * 256 + si) * 32;
    if (j < 32) out[base + j] = val;
    else        out2[base + (j - 32)] = val;
  } else if (mode == 2) {
    out[(size_t)row * N + col] = val;
  } else {
    out[(size_t)row * N + col] =
        0.5f * val * (1.f + erff(val * 0.70710678118654752f));
  }
}

__global__ __launch_bounds__(256) void gemm64x64_f32(
    const float* __restrict__ A0, const float* __restrict__ A1, int K0,
    const float* __restrict__ Wn, const float* __restrict__ bias,
    float* __restrict__ out, float* __restrict__ out2, int M, int N, int K,
    int mode) {
  // stride 68 floats = 272B (16B-multiple for b128 async writes); 16 rows at
  // the same column hit 16 distinct banks (68 == 4 mod 64).
  __shared__ float As[64][68];
  __shared__ float Bs[64][68];  // row-major [n][k] (async-friendly)
  const int tid = threadIdx.x;
  const int wave = tid >> 5, lane = tid & 31;
  const int ln = lane & 15, hi = lane >> 4;
  const int r0 = blockIdx.y * 64, n0 = blockIdx.x * 64;
  const int rt = wave & 3;   // 4 row tiles of 16
  const int cg = wave >> 2;  // 2 col groups of 2 tiles
  const int K1 = K - K0;
  const unsigned ldsA = lds_addr(&As[0][0]);
  const unsigned ldsB = lds_addr(&Bs[0][0]);

  v8f acc0 = zero8(), acc1 = zero8();

  for (int kc = 0; kc < K; kc += 64) {
    // A tile: 64 rows x 64 k, 16B per async op, 4 ops/thread.
    const float* srcA;
    int ldA;
    if (kc < K0) { srcA = A0 + (size_t)r0 * K0 + kc;        ldA = K0; }
    else         { srcA = A1 + (size_t)r0 * K1 + (kc - K0); ldA = K1; }
#pragma unroll
    for (int e = tid; e < 1024; e += 256) {
      int r = e >> 4, c4 = (e & 15) << 2;
      async_b128(ldsA + (unsigned)(r * 68 + c4) * 4, (r * ldA + c4) * 4, srcA);
    }
    // B tile: 64 rows (n) x 64 k.
    const float* srcB = Wn + (size_t)n0 * K + kc;
#pragma unroll
    for (int e = tid; e < 1024; e += 256) {
      int nn = e >> 4, c4 = (e & 15) << 2;
      async_b128(ldsB + (unsigned)(nn * 68 + c4) * 4, (nn * K + c4) * 4, srcB);
    }
    async_wait0();
    __syncthreads();
#pragma unroll
    for (int kk = 0; kk < 64; kk += 4) {
      v2f a, b0, b1;
      a.x = As[rt * 16 + ln][kk + hi * 2];
      a.y = As[rt * 16 + ln][kk + hi * 2 + 1];
      int c0r = (cg * 2) * 16 + ln;
      b0.x = Bs[c0r][kk + hi * 2];
      b0.y = Bs[c0r][kk + hi * 2 + 1];
      b1.x = Bs[c0r + 16][kk + hi * 2];
      b1.y = Bs[c0r + 16][kk + hi * 2 + 1];
      acc0 = wmma4(a, b0, acc0);
      acc1 = wmma4(a, b1, acc1);
    }
    __syncthreads();
  }
#pragma unroll
  for (int i = 0; i < 8; ++i) {
    int row = r0 + rt * 16 + i + hi * 8;
    int c0 = n0 + (cg * 2) * 16 + ln;
    gemm_store(mode, row, c0, acc0[i], bias, out, out2, N);
    gemm_store(mode, row, c0 + 16, acc1[i], bias, out, out2, N);
  }
}

// ---------------------------------------------------------------------------
// L2-normalize rows of 32 (cp), one wave per row.
// ---------------------------------------------------------------------------
__global__ __launch_bounds__(256) void l2n32(float* __restrict__ cp) {
  int wave = threadIdx.x >> 5, lane = threadIdx.x & 31;
  size_t row = (size_t)blockIdx.x * 8 + wave;
  float v = cp[row * 32 + lane];
  float s = v * v;
#pragma unroll
  for (int m = 16; m >= 1; m >>= 1) s += __shfl_xor(s, m, 32);
  cp[row * 32 + lane] = v / fmaxf(sqrtf(s), 1e-12f);
}

// ---------------------------------------------------------------------------
// Similarity + sigmoid + max/argmax + gather-scale, fused.
// Block: 128 threads (4 waves), each wave owns 16 rows; block covers 64 rows
// of l for one head m. K=32 WMMA against all 256 centers.
// ---------------------------------------------------------------------------
__global__ __launch_bounds__(128) void simdisp(
    const float* __restrict__ xp, const float* __restrict__ cpn,
    const float* __restrict__ cv, const float* __restrict__ alpha,
    const float* __restrict__ beta, float* __restrict__ disp) {
  __shared__ float xpl[64][36];   // 144B rows: b128-aligned
  __shared__ float cl[256][36];
  __shared__ float invn[64];
  __shared__ float vmx[64];
  __shared__ int imx[64];
  const int tid = threadIdx.x;
  const int wave = tid >> 5, lane = tid & 31;
  const int ln = lane & 15, hi = lane >> 4;
  const int m = blockIdx.y;
  const int l0 = blockIdx.x * 64;
  const float* xpm = xp + ((size_t)m * 4096 + l0) * 32;
  const float* cm = cpn + (size_t)m * 256 * 32;
  const unsigned ldsX = lds_addr(&xpl[0][0]);
  const unsigned ldsC = lds_addr(&cl[0][0]);

#pragma unroll
  for (int e = tid; e < 512; e += 128) {  // 64 rows x 32 (8 x 16B chunks/row)
    int r = e >> 3, c4 = (e & 7) << 2;
    async_b128(ldsX + (unsigned)(r * 36 + c4) * 4, e << 4, xpm);
  }
  for (int e = tid; e < 2048; e += 128) {  // 256 rows x 32
    int r = e >> 3, c4 = (e & 7) << 2;
    async_b128(ldsC + (unsigned)(r * 36 + c4) * 4, e << 4, cm);
  }
  async_wait0();
  __syncthreads();
  if (tid < 64) {
    float s = 0.f;
#pragma unroll
    for (int k = 0; k < 32; ++k) {
      float v = xpl[tid][k];
      s += v * v;
    }
    invn[tid] = 1.f / fmaxf(sqrtf(s), 1e-12f);
  }
  __syncthreads();

  const float Av = alpha[0], Bv = beta[0];
  const int rt = wave;
  float bv[8];
  int bi[8];
#pragma unroll
  for (int i = 0; i < 8; ++i) {
    bv[i] = -1e30f;
    bi[i] = 0;
  }

  for (int ct = 0; ct < 16; ++ct) {
    v8f acc = zero8();
#pragma unroll
    for (int kk = 0; kk < 32; kk += 4) {
      v2f a, b;
      a.x = xpl[rt * 16 + ln][kk + hi * 2];
      a.y = xpl[rt * 16 + ln][kk + hi * 2 + 1];
      b.x = cl[ct * 16 + ln][kk + hi * 2];
      b.y = cl[ct * 16 + ln][kk + hi * 2 + 1];
      acc = wmma4(a, b, acc);
    }
#pragma unroll
    for (int i = 0; i < 8; ++i) {
      int r = rt * 16 + i + hi * 8;
      float z = Av * (acc[i] * invn[r]) + Bv;
      float t = 1.f / (1.f + expf(-z));
      int col = ct * 16 + ln;
      if (t > bv[i]) {  // strict: keeps first occurrence within ascending cols
        bv[i] = t;
        bi[i] = col;
      }
    }
  }
  // argmax reduce across the 16 lanes holding the 16 cols of each row
#pragma unroll
  for (int i = 0; i < 8; ++i) {
    float v = bv[i];
    int ix = bi[i];
#pragma unroll
    for (int mm = 8; mm >= 1; mm >>= 1) {
      float ov = __shfl_xor(v, mm, 16);
      int oi = __shfl_xor(ix, mm, 16);
      if (ov > v || (ov == v && oi < ix)) {
        v = ov;
        ix = oi;
      }
    }
    if (ln == 0) {
      int r = rt * 16 + i + hi * 8;
      vmx[r] = v;
      imx[r] = ix;
    }
  }
  __syncthreads();

  const int n = m >> 3, f = m & 7;
  for (int e = tid; e < 64 * 32; e += 128) {
    int r = e >> 5, j = e & 31;
    float val = vmx[r] * cv[((size_t)m * 256 + imx[r]) * 32 + j];
    disp[(size_t)(n * 4096 + l0 + r) * 256 + f * 32 + j] = val;
  }
}

// ---------------------------------------------------------------------------
// In-place LayerNorm over 256, one wave per row.
// ---------------------------------------------------------------------------
__global__ __launch_bounds__(256) void ln256(float* __restrict__ x,
                                             const float* __restrict__ g,
                                             const float* __restrict__ b) {
  int wave = threadIdx.x >> 5, lane = threadIdx.x & 31;
  size_t base = ((size_t)blockIdx.x * 8 + wave) * 256;
  float v[8], s = 0.f, s2 = 0.f;
#pragma unroll
  for (int j = 0; j < 8; ++j) {
    v[j] = x[base + lane + j * 32];
    s += v[j];
    s2 += v[j] * v[j];
  }
#pragma unroll
  for (int m = 16; m >= 1; m >>= 1) {
    s += __shfl_xor(s, m, 32);
    s2 += __shfl_xor(s2, m, 32);
  }
  float mu = s * (1.f / 256.f);
  float var = fmaxf(s2 * (1.f / 256.f) - mu * mu, 0.f);
  float rstd = rsqrtf(var + 1e-5f);
#pragma unroll
  for (int j = 0; j < 8; ++j) {
    int ch = lane + j * 32;
    x[base + ch] = (v[j] - mu) * rstd * g[ch] + b[ch];
  }
}

// ---------------------------------------------------------------------------
// Pack conv weights OIHW(256,512,3,3) -> [tap][o=256][k=512] so conv B tiles
// are contiguous-per-row b128 async copies.
// ---------------------------------------------------------------------------
__global__ __launch_bounds__(256) void packw(const float* __restrict__ cw,
                                             float* __restrict__ wp) {
  size_t i = (size_t)blockIdx.x * 256 + threadIdx.x;
  if (i >= (size_t)9 * 512 * 256) return;
  int k = (int)(i & 511);
  int o = (int)((i >> 9) & 255);
  int tap = (int)(i >> 17);
  wp[i] = cw[((size_t)o * 512 + k) * 9 + tap];
}

// ---------------------------------------------------------------------------
// 3x3 conv (implicit GEMM over 9 taps) + bias + LayerNorm(norm1) + residual.
// Block owns 32 contiguous pixels of one image row (uniform halo masks) and
// ALL 256 output channels -> LN fuses in the epilogue (saves a 64MB trip).
// ---------------------------------------------------------------------------
__global__ __launch_bounds__(256) void conv_ln(
    const float* __restrict__ Y, const float* __restrict__ Wp,
    const float* __restrict__ cbias, const float* __restrict__ g,
    const float* __restrict__ bta, const float* __restrict__ x0,
    float* __restrict__ out) {
  __shared__ float As[32][36];    // 32 pixels x 32 k (regular path: halo mask)
  __shared__ float Bs[256][36];   // 256 o x 32 k, 144B rows (async b128)
  const int tid = threadIdx.x, wave = tid >> 5, lane = tid & 31;
  const int ln = lane & 15, hi = lane >> 4;
  const int pb = blockIdx.x;
  const int n = pb >> 7;
  const int rem = pb & 127;
  const int h = rem >> 1;
  const int w0 = (rem & 1) * 32;
  const int rt = wave & 1;             // 2 row tiles of 16 pixels
  const int ctbase = (wave >> 1) * 4;  // 4 col tiles per wave -> 16 total
  const unsigned ldsB = lds_addr(&Bs[0][0]);

  v8f acc[4];
#pragma unroll
  for (int t = 0; t < 4; ++t) acc[t] = zero8();

  for (int tap = 0; tap < 9; ++tap) {
    const int kh = tap / 3, kw = tap % 3;
    const int hh = h + kh - 1;
    if (hh < 0 || hh >= 64) continue;  // uniform across block
    for (int kc = 0; kc < 512; kc += 32) {
      // B tile async: 256 rows (o) x 32 k, 8 x b128 per thread.
      const float* srcB = Wp + (size_t)tap * (256 * 512) + kc;
      __builtin_prefetch(srcB + 32, 0, 0);  // next k-chunk
#pragma unroll
      for (int e = tid; e < 2048; e += 256) {
        int o = e >> 3, c4 = (e & 7) << 2;
        async_b128(ldsB + (unsigned)(o * 36 + c4) * 4, (o * 512 + c4) * 4,
                   srcB);
      }
      // A tile regular (zero-filled halo).
      for (int e = tid; e < 32 * 32; e += 256) {
        int p = e >> 5, ck = e & 31;
        int ww = w0 + p + kw - 1;
        As[p][ck] = (ww >= 0 && ww < 64)
                        ? Y[(size_t)(n * 4096 + hh * 64 + ww) * 512 + kc + ck]
                        : 0.f;
      }
      async_wait0();
      __syncthreads();
#pragma unroll
      for (int kk = 0; kk < 32; kk += 4) {
        v2f a;
        a.x = As[rt * 16 + ln][kk + hi * 2];
        a.y = As[rt * 16 + ln][kk + hi * 2 + 1];
#pragma unroll
        for (int t = 0; t < 4; ++t) {
          v2f b;
          int cc = (ctbase + t) * 16 + ln;
          b.x = Bs[cc][kk + hi * 2];
          b.y = Bs[cc][kk + hi * 2 + 1];
          acc[t] = wmma4(a, b, acc[t]);
        }
      }
      __syncthreads();
    }
  }

  // Stage bias-added conv outputs in LDS (reuse Bs: 32*260 <= 256*36), then
  // fused LN + residual.
  float(*ot)[260] = (float(*)[260]) & Bs[0][0];
#pragma unroll
  for (int t = 0; t < 4; ++t) {
    int cc = (ctbase + t) * 16 + ln;
    float bv = cbias[cc];
#pragma unroll
    for (int i = 0; i < 8; ++i) {
      int p = rt * 16 + i + hi * 8;
      ot[p][cc] = acc[t][i] + bv;
    }
  }
  __syncthreads();

#pragma unroll
  for (int q = 0; q < 4; ++q) {
    int p = wave * 4 + q;
    int l = h * 64 + w0 + p;
    float v[8], s = 0.f, s2 = 0.f;
#pragma unroll
    for (int j = 0; j < 8; ++j) {
      v[j] = ot[p][lane + j * 32];
      s += v[j];
      s2 += v[j] * v[j];
    }
#pragma unroll
    for (int mm = 16; mm >= 1; mm >>= 1) {
      s += __shfl_xor(s, mm, 32);
      s2 += __shfl_xor(s2, mm, 32);
    }
    float mu = s * (1.f / 256.f);
    float var = fmaxf(s2 * (1.f / 256.f) - mu * mu, 0.f);
    float rstd = rsqrtf(var + 1e-5f);
    size_t base = (size_t)(n * 4096 + l) * 256;
#pragma unroll
    for (int j = 0; j < 8; ++j) {
      int ch = lane + j * 32;
      out[base + ch] = (v[j] - mu) * rstd * g[ch] + bta[ch] + x0[base + ch];
    }
  }
}

// ---------------------------------------------------------------------------
extern "C" void kernel_launch(void* const* d_in, const int* in_sizes, int n_in,
                              void* d_out, int out_size, void* d_ws,
                              size_t ws_size, hipStream_t stream) {
  (void)in_sizes; (void)n_in; (void)out_size; (void)ws_size;
  const float* x0      = (const float*)d_in[0];
  const float* center1 = (const float*)d_in[1];
  const float* proj0_w = (const float*)d_in[2];
  const float* proj0_b = (const float*)d_in[3];
  const float* proj1_w = (const float*)d_in[4];
  const float* proj1_b = (const float*)d_in[5];
  const float* merge_w = (const float*)d_in[6];
  const float* merge_b = (const float*)d_in[7];
  const float* alpha   = (const float*)d_in[8];
  const float* beta    = (const float*)d_in[9];
  const float* n0g     = (const float*)d_in[10];
  const float* n0b     = (const float*)d_in[11];
  const float* n1g     = (const float*)d_in[12];
  const float* n1b     = (const float*)d_in[13];
  const float* lin_w   = (const float*)d_in[14];
  const float* lin_b   = (const float*)d_in[15];
  const float* conv_w  = (const float*)d_in[16];
  const float* conv_b  = (const float*)d_in[17];
  float* outf = (float*)d_out;

  // Workspace layout (floats), with liveness-based aliasing:
  //   [0, 17825792)  : xp(8M) | cp(512K) | cv(512K) | disp(8M)   (dead by lin)
  //                    Y(16M) overlays this region for lin->conv
  //   [17825792, ..) : new(8M)
  //   [26214400, ..) : packed conv weights (1.125M)
  // Total ~104.5 MiB.
  float* ws   = (float*)d_ws;
  float* xp   = ws;
  float* cp   = ws + 8388608;
  float* cv   = ws + 8912896;
  float* disp = ws + 9437184;
  float* Y    = ws;  // overlays xp/cp/cv/disp (all dead before lin GEMM)
  float* nw   = ws + 17825792;
  float* Wp   = ws + 26214400;

  // 0) pack conv weights
  packw<<<4608, 256, 0, stream>>>(conv_w, Wp);
  // 1) proj0: xp(n*8, l, 32) = x0 @ proj0_w^T + b (scattered)
  gemm64x64_f32<<<dim3(4, 512), 256, 0, stream>>>(
      x0, x0, 256, proj0_w, proj0_b, xp, nullptr, 32768, 256, 256, 0);
  // 2) proj1 -> cp/cv split
  gemm64x64_f32<<<dim3(8, 32), 256, 0, stream>>>(
      center1, center1, 256, proj1_w, proj1_b, cp, cv, 2048, 512, 256, 1);
  // 3) l2-normalize cp rows
  l2n32<<<2048, 256, 0, stream>>>(cp);
  // 4) similarity + sigmoid + argmax + gather -> disp (n,l,256)
  simdisp<<<dim3(64, 64), 128, 0, stream>>>(xp, cp, cv, alpha, beta, disp);
  // 5) merge GEMM -> new
  gemm64x64_f32<<<dim3(4, 512), 256, 0, stream>>>(
      disp, disp, 256, merge_w, merge_b, nw, nullptr, 32768, 256, 256, 2);
  // 6) LayerNorm(norm0) in-place
  ln256<<<4096, 256, 0, stream>>>(nw, n0g, n0b);
  // 7) lin GEMM on virtual concat [x0 | new] + exact GELU -> Y (NHWC, 512ch)
  gemm64x64_f32<<<dim3(8, 512), 256, 0, stream>>>(
      x0, nw, 256, lin_w, lin_b, Y, nullptr, 32768, 512, 512, 3);
  // 8) conv3x3 + bias + LayerNorm(norm1) + residual -> out
  conv_ln<<<1024, 256, 0, stream>>>(Y, Wp, conv_b, n1g, n1b, x0, outf);
}